// GCPLoss_53291954208910
// MI455X (gfx1250) — compile-verified
//
#include <hip/hip_runtime.h>
#include <hip/hip_bf16.h>

// ---------------------------------------------------------------------------
// GCPLoss for MI455X (gfx1250, wave32).
//
//  chamfer : B=2, N=M=8192 pairwise sq-dist, dual min + mean -> WMMA f32 16x16x4
//            (each wave owns 32 rows = 2 A-tiles; one B tile + one C broadcast
//             feeds 2 WMMAs per j-step to halve VALU overhead per matrix op;
//             512 waves keep the WGP array filled)
//  cldice  : 3x3x3 min-pool erosion on [2,1,64,128,128] -> bandwidth kernel
//  depth   : SILog + grad-L1 + masked-L1 on [2,512,512]  -> bandwidth kernel
//
// ws layout (floats):
//   [0      .. 16384) rowmin   init +inf
//   [16384  .. 32768) colmin   init +inf
//   [32768  .. 32774) cldice acc: inter[2], psum[2], gsum[2]          init 0
//   [32774  .. 32782) depth  acc: sg[2], sg2[2], sx, sy, mnum, mden   init 0
// ---------------------------------------------------------------------------

typedef __attribute__((ext_vector_type(2))) float v2f;
typedef __attribute__((ext_vector_type(8))) float v8f;

#define BATCH   2
#define NPTS    8192          // N == M
#define VD      64
#define VH      128
#define VW      128
#define DHW     (VD*VH*VW)    // 1048576
#define DH      512
#define DW      512
#define DPIX    (DH*DW)       // 262144

#define ROWTILES 2            // 16-row tiles per wave (32 rows/wave)

#define WS_ROWMIN  0
#define WS_COLMIN  16384
#define WS_ACCC    32768
#define WS_ACCD    32774
#define WS_TOTAL   32782

// ---------------------------------------------------------------- init
__global__ void gcp_init_kernel(float* ws) {
    int i = blockIdx.x * blockDim.x + threadIdx.x;
    if (i < WS_ACCC) ws[i] = INFINITY;          // rowmin + colmin
    else if (i < WS_TOTAL) ws[i] = 0.0f;        // accumulators
}

// ---------------------------------------------------------------- chamfer
// D = (-2*A) * B + C with C[i][j] = |b_j|^2  ->  D[i][j] = g2_j - 2*dot(a_i,b_j)
// min over j kept per lane, |a_i|^2 added after the reduction.
// A (16x4 f32) lane layout: lanes 0-15 -> {x,y} of point (lane&15),
//                           lanes 16-31 -> {z,0}.  B (4x16) mirrors it.
__global__ void chamfer_min_kernel(const float* __restrict__ rows_pts,
                                   const float* __restrict__ cols_pts,
                                   float* __restrict__ out_min) {
    const int lane = threadIdx.x & 31;
    const int wave = blockIdx.x * (blockDim.x >> 5) + (threadIdx.x >> 5);
    const int row0 = wave * (16 * ROWTILES);    // global row index over B*NPTS
    const int b    = row0 / NPTS;
    const int r0   = row0 - b * NPTS;           // row base within batch
    const float* __restrict__ Ab = rows_pts + (size_t)b * NPTS * 3;
    const float* __restrict__ Bb = cols_pts + (size_t)b * NPTS * 3;

    // A operands (scaled by -2), one per 16-row tile
    v2f a[ROWTILES];
#pragma unroll
    for (int t = 0; t < ROWTILES; ++t) {
        const int pi = r0 + t * 16 + (lane & 15);
        const float ax = Ab[pi * 3 + 0];
        const float ay = Ab[pi * 3 + 1];
        const float az = Ab[pi * 3 + 2];
        a[t].x = (lane < 16) ? (-2.0f * ax) : (-2.0f * az);
        a[t].y = (lane < 16) ? (-2.0f * ay) : 0.0f;
    }

    float m[ROWTILES][8];
#pragma unroll
    for (int t = 0; t < ROWTILES; ++t)
#pragma unroll
        for (int v = 0; v < 8; ++v) m[t][v] = INFINITY;

    for (int j0 = 0; j0 < NPTS; j0 += 16) {
        const int qi = j0 + (lane & 15);
        const float bx = Bb[qi * 3 + 0];
        const float by = Bb[qi * 3 + 1];
        const float bz = Bb[qi * 3 + 2];
        const float g2 = bx * bx + by * by + bz * bz;
        v2f bm;
        bm.x = (lane < 16) ? bx : bz;
        bm.y = (lane < 16) ? by : 0.0f;
        v8f c;
#pragma unroll
        for (int v = 0; v < 8; ++v) c[v] = g2;   // C[i][j] = |g_j|^2 (j = lane&15)
        // shared B tile + shared C broadcast feed ROWTILES WMMAs
#pragma unroll
        for (int t = 0; t < ROWTILES; ++t) {
            // 8 args: (neg_a, A, neg_b, B, c_mod, C, reuse_a, reuse_b)
            v8f d = __builtin_amdgcn_wmma_f32_16x16x4_f32(
                false, a[t], false, bm, (short)0, c, false, false);
#pragma unroll
            for (int v = 0; v < 8; ++v) m[t][v] = fminf(m[t][v], d[v]);
        }
    }

    // Butterfly min across each 16-lane half, then write per-row minima.
#pragma unroll
    for (int t = 0; t < ROWTILES; ++t) {
#pragma unroll
        for (int v = 0; v < 8; ++v) {
            float x = m[t][v];
            x = fminf(x, __shfl_xor(x, 1, 32));
            x = fminf(x, __shfl_xor(x, 2, 32));
            x = fminf(x, __shfl_xor(x, 4, 32));
            x = fminf(x, __shfl_xor(x, 8, 32));
            m[t][v] = x;
        }
        // lane v (half 0) -> row base+v ; lane 16+v (half 1) -> row base+8+v
#pragma unroll
        for (int v = 0; v < 8; ++v) {
            if ((lane & 15) == v) {
                const int i  = v + ((lane >= 16) ? 8 : 0);
                const int gi = r0 + t * 16 + i;
                const float px = Ab[gi * 3 + 0];
                const float py = Ab[gi * 3 + 1];
                const float pz = Ab[gi * 3 + 2];
                out_min[(size_t)b * NPTS + gi] =
                    m[t][v] + px * px + py * py + pz * pz;
            }
        }
    }
}

// ---------------------------------------------------------------- wave reduce
__device__ __forceinline__ float wave_sum(float x) {
    x += __shfl_xor(x, 1,  32);
    x += __shfl_xor(x, 2,  32);
    x += __shfl_xor(x, 4,  32);
    x += __shfl_xor(x, 8,  32);
    x += __shfl_xor(x, 16, 32);
    return x;
}

// ---------------------------------------------------------------- cldice
__global__ void cldice_kernel(const float* __restrict__ pv,
                              const float* __restrict__ gv,
                              float* __restrict__ acc) {   // inter[2],psum[2],gsum[2]
    const int t = blockIdx.x * blockDim.x + threadIdx.x;   // grid == B*DHW exactly
    const int b = t / DHW;
    const int r = t - b * DHW;
    const int z  = r / (VH * VW);
    const int rem = r - z * (VH * VW);
    const int y  = rem / VW;
    const int x  = rem - y * VW;
    const float* __restrict__ p = pv + (size_t)b * DHW;
    const float* __restrict__ g = gv + (size_t)b * DHW;

    float mp = INFINITY, mg = INFINITY;
#pragma unroll
    for (int dz = -1; dz <= 1; ++dz) {
        int zz = z + dz; zz = (zz < 0) ? 0 : ((zz > VD - 1) ? VD - 1 : zz);
#pragma unroll
        for (int dy = -1; dy <= 1; ++dy) {
            int yy = y + dy; yy = (yy < 0) ? 0 : ((yy > VH - 1) ? VH - 1 : yy);
#pragma unroll
            for (int dx = -1; dx <= 1; ++dx) {
                int xx = x + dx; xx = (xx < 0) ? 0 : ((xx > VW - 1) ? VW - 1 : xx);
                const int o = (zz * VH + yy) * VW + xx;
                const float sp = 1.0f / (1.0f + __expf(-p[o]));  // sigmoid
                mp = fminf(mp, sp);
                mg = fminf(mg, g[o]);
            }
        }
    }
    float s_inter = wave_sum(mp * mg);
    float s_p     = wave_sum(mp);
    float s_g     = wave_sum(mg);
    if ((threadIdx.x & 31) == 0) {
        atomicAdd(&acc[0 + b], s_inter);
        atomicAdd(&acc[2 + b], s_p);
        atomicAdd(&acc[4 + b], s_g);
    }
}

// ---------------------------------------------------------------- depth loss
__global__ void depth_kernel(const float* __restrict__ pred,
                             const float* __restrict__ gt,
                             const float* __restrict__ mask,
                             float* __restrict__ acc) { // sg[2],sg2[2],sx,sy,mnum,mden
    const int t = blockIdx.x * blockDim.x + threadIdx.x;  // grid == B*DPIX exactly
    const int b = t / DPIX;
    const int r = t - b * DPIX;
    const int y = r / DW;
    const int x = r - y * DW;
    const size_t base = (size_t)b * DPIX;
    const float pv = pred[base + r];
    const float gv = gt[base + r];

    const float gval = logf(pv + 0.1f) - logf(gv + 0.1f);

    float tx = 0.0f, ty = 0.0f;
    if (y < DH - 1) {
        const float pgx = fabsf(pv - pred[base + r + DW]);
        const float ggx = fabsf(gv - gt[base + r + DW]);
        tx = fabsf(pgx - ggx);
    }
    if (x < DW - 1) {
        const float pgy = fabsf(pv - pred[base + r + 1]);
        const float ggy = fabsf(gv - gt[base + r + 1]);
        ty = fabsf(pgy - ggy);
    }
    const float mk = mask[base + r];
    const float ml = fabsf(pv - gv) * mk;

    const float sg  = wave_sum(gval);
    const float sg2 = wave_sum(gval * gval);
    const float sx  = wave_sum(tx);
    const float sy  = wave_sum(ty);
    const float mn  = wave_sum(ml);
    const float md  = wave_sum(mk);
    if ((threadIdx.x & 31) == 0) {
        atomicAdd(&acc[0 + b], sg);
        atomicAdd(&acc[2 + b], sg2);
        atomicAdd(&acc[4], sx);
        atomicAdd(&acc[5], sy);
        atomicAdd(&acc[6], mn);
        atomicAdd(&acc[7], md);
    }
}

// ---------------------------------------------------------------- combine
__global__ void combine_kernel(const float* __restrict__ ws,
                               const int* __restrict__ iter_ptr,
                               float* __restrict__ out) {
    __shared__ float sA[256];
    __shared__ float sB[256];
    float la = 0.0f, lb = 0.0f;
    for (int i = threadIdx.x; i < BATCH * NPTS; i += 256) {
        la += ws[WS_ROWMIN + i];
        lb += ws[WS_COLMIN + i];
    }
    sA[threadIdx.x] = la;
    sB[threadIdx.x] = lb;
    __syncthreads();
    for (int s = 128; s > 0; s >>= 1) {
        if (threadIdx.x < s) {
            sA[threadIdx.x] += sA[threadIdx.x + s];
            sB[threadIdx.x] += sB[threadIdx.x + s];
        }
        __syncthreads();
    }
    if (threadIdx.x == 0) {
        const float invBN = 1.0f / (float)(BATCH * NPTS);
        const float chamfer = sA[0] * invBN + sB[0] * invBN;

        const float* aC = ws + WS_ACCC;
        const float smooth = 1e-5f;
        float dsum = 0.0f;
        for (int b = 0; b < BATCH; ++b)
            dsum += (2.0f * aC[b] + smooth) / (aC[2 + b] + aC[4 + b] + smooth);
        const float cldice = 1.0f - dsum / (float)BATCH;

        const float* aD = ws + WS_ACCD;
        const float P = (float)DPIX;
        float vsum = 0.0f, msum = 0.0f;
        for (int b = 0; b < BATCH; ++b) {
            const float mg  = aD[b] / P;
            const float var = aD[2 + b] / P - mg * mg;
            vsum += var;
            msum += mg * mg;
        }
        const float silog = 10.0f * (vsum / (float)BATCH) + 10.0f * (msum / (float)BATCH);
        const float grad  = aD[4] / (float)(BATCH * (DH - 1) * DW)
                          + aD[5] / (float)(BATCH * DH * (DW - 1));
        const float ml1   = aD[6] / (aD[7] + 1e-8f);
        const float dloss = silog + grad + ml1;

        int it = *iter_ptr;
        if (it < 1) it = 1;
        const float gamma1 = 2.0f * logf((float)it / 20000.0f);
        out[0] = gamma1 * chamfer + 0.5f * cldice + 0.01f * dloss;
    }
}

// ---------------------------------------------------------------- launch
extern "C" void kernel_launch(void* const* d_in, const int* in_sizes, int n_in,
                              void* d_out, int out_size, void* d_ws, size_t ws_size,
                              hipStream_t stream) {
    const float* pred_centers = (const float*)d_in[0];
    const float* pred_volume  = (const float*)d_in[1];
    const float* pred_depth   = (const float*)d_in[2];
    const float* gt_points    = (const float*)d_in[3];
    const float* gt_volume    = (const float*)d_in[4];
    const float* gt_depth     = (const float*)d_in[5];
    const float* depth_mask   = (const float*)d_in[6];
    const int*   iteration    = (const int*)d_in[7];
    float* ws  = (float*)d_ws;
    float* out = (float*)d_out;

    // init workspace (mins -> +inf, accumulators -> 0)
    gcp_init_kernel<<<(WS_TOTAL + 255) / 256, 256, 0, stream>>>(ws);

    // chamfer: 32 rows per wave, 8 waves per block -> 512 waves per direction
    const int waves  = (BATCH * NPTS) / (16 * ROWTILES);   // 512
    const int blocks = waves / 8;                          // 64
    chamfer_min_kernel<<<blocks, 256, 0, stream>>>(pred_centers, gt_points,
                                                   ws + WS_ROWMIN);
    chamfer_min_kernel<<<blocks, 256, 0, stream>>>(gt_points, pred_centers,
                                                   ws + WS_COLMIN);

    // cldice erosion + sums
    cldice_kernel<<<(BATCH * DHW) / 256, 256, 0, stream>>>(pred_volume, gt_volume,
                                                           ws + WS_ACCC);

    // depth loss sums
    depth_kernel<<<(BATCH * DPIX) / 256, 256, 0, stream>>>(pred_depth, gt_depth,
                                                           depth_mask, ws + WS_ACCD);

    // final scalar
    combine_kernel<<<1, 256, 0, stream>>>(ws, iteration, out);
}